// MambaLayer_81655918232060
// MI455X (gfx1250) — compile-verified
//
#include <hip/hip_runtime.h>

// ---------------------------------------------------------------------------
// Problem constants (from the reference)
// ---------------------------------------------------------------------------
#define BB      2          // batch
#define NPART   3          // x1/x2/x3
#define CDIM    128        // DIM
#define DI      256        // D_INNER
#define DS      16         // D_STATE
#define NTOK    4096       // 16^3
#define LSEQ    (NPART*NTOK)   // 12288 tokens per mamba sequence
#define DBLW    48         // padded width of xproj output (dt8 | B16 | C16 | pad8)
#define NCH     96         // scan chunks
#define CLEN    128        // chunk length (96*128 = 12288)

typedef float v2f __attribute__((ext_vector_type(2)));
typedef float v8f __attribute__((ext_vector_type(8)));

// fp32 WMMA, D(16x16) = A(16x4) * B(4x16) + C  -> v_wmma_f32_16x16x4_f32
static __device__ __forceinline__ v8f wmma4(v2f a, v2f b, v8f c) {
  return __builtin_amdgcn_wmma_f32_16x16x4_f32(false, a, false, b, (short)0, c,
                                               false, false);
}

static __device__ __forceinline__ float sigmoidf_(float x) {
  return 1.f / (1.f + __expf(-x));
}
static __device__ __forceinline__ float siluf_(float x) { return x * sigmoidf_(x); }

// branch permutation: scan position t -> mamba sequence position l
//   br 0: forward, br 1: reversed, br 2: slice-interleaved (NSLICES=16)
static __device__ __forceinline__ int sigma_map(int br, int t) {
  if (br == 0) return t;
  if (br == 1) return LSEQ - 1 - t;
  return (t & 15) * (LSEQ / 16) + (t >> 4);
}

// ---------------------------------------------------------------------------
// out = x  (residual base; layout of x and out is identical)
// ---------------------------------------------------------------------------
__global__ void k_copy(const float* __restrict__ src, float* __restrict__ dst, int n) {
  int i = blockIdx.x * blockDim.x + threadIdx.x;
  if (i < n) dst[i] = src[i];
}

// ---------------------------------------------------------------------------
// LayerNorm over C=128 for each of B*3*4096 tokens; one wave32 per token.
// normed layout: [b][part][tok][c]  (c contiguous)
// ---------------------------------------------------------------------------
__global__ void k_layernorm(const float* __restrict__ x, const float* __restrict__ w,
                            const float* __restrict__ bias, float* __restrict__ normed) {
  int wid  = (blockIdx.x * blockDim.x + threadIdx.x) >> 5;
  int lane = threadIdx.x & 31;
  if (wid >= BB * NPART * NTOK) return;
  int b    = wid / (NPART * NTOK);
  int rem  = wid % (NPART * NTOK);
  int part = rem / NTOK;
  int tok  = rem % NTOK;
  const float* xp = x + ((size_t)b * (NPART * CDIM) + part * CDIM) * NTOK + tok;
  float v[4];
  float s = 0.f, s2 = 0.f;
#pragma unroll
  for (int i = 0; i < 4; ++i) {
    v[i] = xp[(size_t)(lane * 4 + i) * NTOK];
    s += v[i];
    s2 += v[i] * v[i];
  }
#pragma unroll
  for (int m = 16; m >= 1; m >>= 1) {
    s  += __shfl_xor(s,  m, 32);
    s2 += __shfl_xor(s2, m, 32);
  }
  float mean = s * (1.f / CDIM);
  float var  = s2 * (1.f / CDIM) - mean * mean;
  float inv  = rsqrtf(var + 1e-5f);
  float* op = normed + (((size_t)b * NPART + part) * NTOK + tok) * CDIM;
#pragma unroll
  for (int i = 0; i < 4; ++i) {
    int c = lane * 4 + i;
    op[c] = (v[i] - mean) * inv * w[c] + bias[c];
  }
}

// ---------------------------------------------------------------------------
// in_proj GEMM (WMMA): xz[b][l][e] = sum_d seq[b,l,d] * W[e,d]
// seq is gathered from normed: blocked (m=0) or interleaved (m=1) ordering.
// 2x2 register blocking: one wave computes a 32x32 output super-tile, so each
// A/B fragment load feeds two WMMAs (1 load : 1 wmma), halving L2 tile traffic.
// ---------------------------------------------------------------------------
__global__ void k_inproj(const float* __restrict__ normed, const float* __restrict__ W,
                         float* __restrict__ xz, int interleaved) {
  int w    = (blockIdx.x * blockDim.x + threadIdx.x) >> 5;
  int lane = threadIdx.x & 31;
  const int NT = (2 * DI) / 32;  // 16 column super-tiles
  if (w >= (BB * LSEQ / 32) * NT) return;
  int tm = w / NT, tn = w % NT;
  int b  = tm / (LSEQ / 32);
  int l0 = (tm % (LSEQ / 32)) * 32;
  int lr = lane & 15, hi = lane >> 4;
  int row0 = l0 + lr, row1 = row0 + 16;
  int p0, q0, p1, q1;
  if (interleaved) { p0 = row0 % 3; q0 = row0 / 3; p1 = row1 % 3; q1 = row1 / 3; }
  else { p0 = row0 / NTOK; q0 = row0 % NTOK; p1 = row1 / NTOK; q1 = row1 % NTOK; }
  const float* a0p = normed + (((size_t)b * NPART + p0) * NTOK + q0) * CDIM + 2 * hi;
  const float* a1p = normed + (((size_t)b * NPART + p1) * NTOK + q1) * CDIM + 2 * hi;
  int e0 = tn * 32 + lr, e1 = e0 + 16;
  const float* b0p = W + (size_t)e0 * CDIM + 2 * hi;
  const float* b1p = W + (size_t)e1 * CDIM + 2 * hi;
  v8f acc00 = {}, acc01 = {}, acc10 = {}, acc11 = {};
#pragma unroll 2
  for (int ko = 0; ko < CDIM; ko += 4) {
    v2f a0 = *(const v2f*)(a0p + ko);
    v2f a1 = *(const v2f*)(a1p + ko);
    v2f b0 = *(const v2f*)(b0p + ko);
    v2f b1 = *(const v2f*)(b1p + ko);
    acc00 = wmma4(a0, b0, acc00);
    acc01 = wmma4(a0, b1, acc01);
    acc10 = wmma4(a1, b0, acc10);
    acc11 = wmma4(a1, b1, acc11);
  }
#pragma unroll
  for (int i = 0; i < 8; ++i) {
    int tk0 = l0 + i + 8 * hi;
    int tk1 = tk0 + 16;
    size_t r0 = (((size_t)b * LSEQ) + tk0) * (2 * DI);
    size_t r1 = (((size_t)b * LSEQ) + tk1) * (2 * DI);
    xz[r0 + e0] = acc00[i];
    xz[r0 + e1] = acc01[i];
    xz[r1 + e0] = acc10[i];
    xz[r1 + e1] = acc11[i];
  }
}

// ---------------------------------------------------------------------------
// Causal depthwise conv (k=4) over the branch-permuted sequence + SiLU.
// u layout: [b][t][d] (scan order)
// ---------------------------------------------------------------------------
__global__ void k_conv(const float* __restrict__ xz, const float* __restrict__ cw,
                       const float* __restrict__ cb, float* __restrict__ u, int br) {
  size_t g = (size_t)blockIdx.x * blockDim.x + threadIdx.x;
  if (g >= (size_t)BB * LSEQ * DI) return;
  int d = (int)(g % DI);
  int t = (int)((g / DI) % LSEQ);
  int b = (int)(g / ((size_t)DI * LSEQ));
  float accv = cb[d];
#pragma unroll
  for (int k = 0; k < 4; ++k) {
    int tt = t - 3 + k;
    if (tt >= 0) {
      int l = sigma_map(br, tt);
      accv += cw[d * 4 + k] * xz[(((size_t)b * LSEQ) + l) * (2 * DI) + d];
    }
  }
  u[(((size_t)b * LSEQ) + t) * DI + d] = siluf_(accv);
}

// ---------------------------------------------------------------------------
// xproj GEMM (WMMA): dbl[b][t][j] = sum_d u[b,t,d] * xproj_w[j,d], j<40 (pad 48)
// ---------------------------------------------------------------------------
__global__ void k_xproj(const float* __restrict__ u, const float* __restrict__ W,
                        float* __restrict__ dbl) {
  int w    = (blockIdx.x * blockDim.x + threadIdx.x) >> 5;
  int lane = threadIdx.x & 31;
  const int NT = 3;
  if (w >= (BB * LSEQ / 16) * NT) return;
  int tm = w / NT, tn = w % NT;
  int b  = tm / (LSEQ / 16);
  int l0 = (tm % (LSEQ / 16)) * 16;
  int lr = lane & 15, hi = lane >> 4;
  int row = l0 + lr;
  const float* arow = u + (((size_t)b * LSEQ) + row) * DI + 2 * hi;
  int j = tn * 16 + lr;
  bool valid = (j < 40);
  const float* brow = W + (size_t)(valid ? j : 0) * DI + 2 * hi;
  v8f acc = {};
  for (int ko = 0; ko < DI; ko += 4) {
    v2f a  = *(const v2f*)(arow + ko);
    v2f bm = {0.f, 0.f};
    if (valid) bm = *(const v2f*)(brow + ko);
    acc = wmma4(a, bm, acc);
  }
#pragma unroll
  for (int i = 0; i < 8; ++i) {
    int tk = l0 + i + 8 * hi;
    dbl[(((size_t)b * LSEQ) + tk) * DBLW + j] = acc[i];
  }
}

// ---------------------------------------------------------------------------
// dtproj GEMM (WMMA, K=8) + bias + softplus -> delta[b][t][d]
// 1x2 N-blocking: each A fragment feeds two WMMAs.
// ---------------------------------------------------------------------------
__global__ void k_dtproj(const float* __restrict__ dbl, const float* __restrict__ W,
                         const float* __restrict__ bias, float* __restrict__ delta) {
  int w    = (blockIdx.x * blockDim.x + threadIdx.x) >> 5;
  int lane = threadIdx.x & 31;
  const int NT = DI / 32;  // 8 column super-tiles
  if (w >= (BB * LSEQ / 16) * NT) return;
  int tm = w / NT, tn = w % NT;
  int b  = tm / (LSEQ / 16);
  int l0 = (tm % (LSEQ / 16)) * 16;
  int lr = lane & 15, hi = lane >> 4;
  int row = l0 + lr;
  const float* arow = dbl + (((size_t)b * LSEQ) + row) * DBLW + 2 * hi;
  int d0 = tn * 32 + lr, d1 = d0 + 16;
  const float* b0p = W + (size_t)d0 * 8 + 2 * hi;
  const float* b1p = W + (size_t)d1 * 8 + 2 * hi;
  v8f acc0 = {}, acc1 = {};
#pragma unroll
  for (int ko = 0; ko < 8; ko += 4) {
    v2f a  = *(const v2f*)(arow + ko);
    v2f b0 = *(const v2f*)(b0p + ko);
    v2f b1 = *(const v2f*)(b1p + ko);
    acc0 = wmma4(a, b0, acc0);
    acc1 = wmma4(a, b1, acc1);
  }
  float bv0 = bias[d0], bv1 = bias[d1];
#pragma unroll
  for (int i = 0; i < 8; ++i) {
    int tk = l0 + i + 8 * hi;
    size_t rbase = (((size_t)b * LSEQ) + tk) * DI;
    float x0 = acc0[i] + bv0;
    float x1 = acc1[i] + bv1;
    delta[rbase + d0] = (x0 > 20.f) ? x0 : log1pf(__expf(x0));
    delta[rbase + d1] = (x1 > 20.f) ? x1 : log1pf(__expf(x1));
  }
}

// ---------------------------------------------------------------------------
// Chunked parallel selective scan.
//   h_t = a_t*h_{t-1} + g_t,  a_t = exp(delta*A[d,n]), g_t = delta*u*B[t,n]
// Pass 1: per-chunk (P = prod a, S = local scan end with h0=0)
// ---------------------------------------------------------------------------
__global__ void k_scan1(const float* __restrict__ delta, const float* __restrict__ u,
                        const float* __restrict__ dbl, const float* __restrict__ A_log,
                        float* __restrict__ P, float* __restrict__ S) {
  int n    = threadIdx.x & 15;
  int dloc = threadIdx.x >> 4;
  int d = blockIdx.y * 16 + dloc;
  int c = blockIdx.x, b = blockIdx.z;
  float Ad = -__expf(A_log[d * DS + n]);
  float p = 1.f, s = 0.f;
  int t0 = c * CLEN;
  for (int tt = 0; tt < CLEN; ++tt) {
    size_t base = (size_t)b * LSEQ + (t0 + tt);
    float dl = delta[base * DI + d];
    float uv = u[base * DI + d];
    float Bt = dbl[base * DBLW + 8 + n];
    float a  = __expf(dl * Ad);
    s = s * a + dl * uv * Bt;
    p *= a;
  }
  size_t idx = (((size_t)b * NCH + c) * DI + d) * DS + n;
  P[idx] = p;
  S[idx] = s;
}

// Pass 2: serial prefix over the 96 chunk summaries (one thread per b,d,n)
__global__ void k_scan2(const float* __restrict__ P, const float* __restrict__ S,
                        float* __restrict__ H0) {
  int g = blockIdx.x * blockDim.x + threadIdx.x;
  if (g >= BB * DI * DS) return;
  int b  = g / (DI * DS);
  int dn = g % (DI * DS);
  float h = 0.f;
  for (int c = 0; c < NCH; ++c) {
    size_t idx = ((size_t)b * NCH + c) * (DI * DS) + dn;
    H0[idx] = h;
    h = S[idx] + P[idx] * h;
  }
}

// Pass 3: re-scan seeded with H0; fuse y = (h.C + u*Dskip) * silu(z), scatter
// through sigma to accumulate into ysum[b][l][d]. 16-lane shuffle reduce over n.
__global__ void k_scan3(const float* __restrict__ delta, const float* __restrict__ u,
                        const float* __restrict__ dbl, const float* __restrict__ xz,
                        const float* __restrict__ A_log, const float* __restrict__ Dskip,
                        const float* __restrict__ H0, float* __restrict__ ysum,
                        int br, int accflag) {
  int n    = threadIdx.x & 15;
  int dloc = threadIdx.x >> 4;
  int d = blockIdx.y * 16 + dloc;
  int c = blockIdx.x, b = blockIdx.z;
  float Ad  = -__expf(A_log[d * DS + n]);
  float dsk = Dskip[d];
  float h = H0[(((size_t)b * NCH + c) * DI + d) * DS + n];
  int t0 = c * CLEN;
  for (int tt = 0; tt < CLEN; ++tt) {
    int t = t0 + tt;
    size_t base = (size_t)b * LSEQ + t;
    float dl = delta[base * DI + d];
    float uv = u[base * DI + d];
    float Bt = dbl[base * DBLW + 8 + n];
    float Ct = dbl[base * DBLW + 24 + n];
    h = h * __expf(dl * Ad) + dl * uv * Bt;
    float contrib = h * Ct;
    contrib += __shfl_xor(contrib, 1, 32);
    contrib += __shfl_xor(contrib, 2, 32);
    contrib += __shfl_xor(contrib, 4, 32);
    contrib += __shfl_xor(contrib, 8, 32);
    if (n == 0) {
      int l = sigma_map(br, t);
      size_t obase = (size_t)b * LSEQ + l;
      float zv = xz[obase * (2 * DI) + DI + d];
      float yv = (contrib + uv * dsk) * siluf_(zv);
      float* dst = ysum + obase * DI + d;
      if (accflag) *dst += yv; else *dst = yv;
    }
  }
}

// ---------------------------------------------------------------------------
// out_proj GEMM (WMMA): out[b][p*128+cc][tok] += sum_d ysum[b,l,d]*W[cc,d]
// (l = p*4096+tok blocked split, identical for both mamba calls per reference)
// 2x2 register blocking like k_inproj.
// ---------------------------------------------------------------------------
__global__ void k_outproj(const float* __restrict__ ysum, const float* __restrict__ W,
                          float* __restrict__ out) {
  int w    = (blockIdx.x * blockDim.x + threadIdx.x) >> 5;
  int lane = threadIdx.x & 31;
  const int NT = CDIM / 32;  // 4 column super-tiles
  if (w >= (BB * LSEQ / 32) * NT) return;
  int tm = w / NT, tn = w % NT;
  int b  = tm / (LSEQ / 32);
  int l0 = (tm % (LSEQ / 32)) * 32;
  int lr = lane & 15, hi = lane >> 4;
  int row0 = l0 + lr, row1 = row0 + 16;
  const float* a0p = ysum + (((size_t)b * LSEQ) + row0) * DI + 2 * hi;
  const float* a1p = ysum + (((size_t)b * LSEQ) + row1) * DI + 2 * hi;
  int c0 = tn * 32 + lr, c1 = c0 + 16;
  const float* b0p = W + (size_t)c0 * DI + 2 * hi;
  const float* b1p = W + (size_t)c1 * DI + 2 * hi;
  v8f acc00 = {}, acc01 = {}, acc10 = {}, acc11 = {};
#pragma unroll 2
  for (int ko = 0; ko < DI; ko += 4) {
    v2f a0 = *(const v2f*)(a0p + ko);
    v2f a1 = *(const v2f*)(a1p + ko);
    v2f b0 = *(const v2f*)(b0p + ko);
    v2f b1 = *(const v2f*)(b1p + ko);
    acc00 = wmma4(a0, b0, acc00);
    acc01 = wmma4(a0, b1, acc01);
    acc10 = wmma4(a1, b0, acc10);
    acc11 = wmma4(a1, b1, acc11);
  }
#pragma unroll
  for (int i = 0; i < 8; ++i) {
    int la = l0 + i + 8 * hi;
    int lb = la + 16;
    int pa = la / NTOK, ta = la % NTOK;
    int pb = lb / NTOK, tb = lb % NTOK;
    size_t ia0 = (((size_t)b * (NPART * CDIM)) + pa * CDIM + c0) * NTOK + ta;
    size_t ia1 = (((size_t)b * (NPART * CDIM)) + pa * CDIM + c1) * NTOK + ta;
    size_t ib0 = (((size_t)b * (NPART * CDIM)) + pb * CDIM + c0) * NTOK + tb;
    size_t ib1 = (((size_t)b * (NPART * CDIM)) + pb * CDIM + c1) * NTOK + tb;
    out[ia0] += acc00[i];
    out[ia1] += acc01[i];
    out[ib0] += acc10[i];
    out[ib1] += acc11[i];
  }
}

// ---------------------------------------------------------------------------
extern "C" void kernel_launch(void* const* d_in, const int* in_sizes, int n_in,
                              void* d_out, int out_size, void* d_ws, size_t ws_size,
                              hipStream_t stream) {
  (void)in_sizes; (void)n_in; (void)out_size; (void)ws_size;
  const float* x          = (const float*)d_in[0];
  const float* norm_w     = (const float*)d_in[1];
  const float* norm_b     = (const float*)d_in[2];
  const float* in_proj_w  = (const float*)d_in[3];
  const float* out_proj_w = (const float*)d_in[4];
  const float *cw[3], *cb[3], *xw[3], *dtw[3], *dtb[3], *alog[3], *dskip[3];
  for (int i = 0; i < 3; ++i) {
    int o = 5 + 7 * i;
    cw[i]    = (const float*)d_in[o + 0];
    cb[i]    = (const float*)d_in[o + 1];
    xw[i]    = (const float*)d_in[o + 2];
    dtw[i]   = (const float*)d_in[o + 3];
    dtb[i]   = (const float*)d_in[o + 4];
    alog[i]  = (const float*)d_in[o + 5];
    dskip[i] = (const float*)d_in[o + 6];
  }
  float* out = (float*)d_out;
  float* ws  = (float*)d_ws;

  size_t off = 0;
  float* normed = ws + off; off += (size_t)BB * NPART * NTOK * CDIM;   // 3.1M
  float* xz     = ws + off; off += (size_t)BB * LSEQ * 2 * DI;        // 12.6M
  float* ubuf   = ws + off; off += (size_t)BB * LSEQ * DI;            // 6.3M
  float* dlt    = ws + off; off += (size_t)BB * LSEQ * DI;            // 6.3M
  float* dbl    = ws + off; off += (size_t)BB * LSEQ * DBLW;          // 1.2M
  float* ysum   = ws + off; off += (size_t)BB * LSEQ * DI;            // 6.3M
  float* Pb     = ws + off; off += (size_t)BB * NCH * DI * DS;        // 0.8M
  float* Sb     = ws + off; off += (size_t)BB * NCH * DI * DS;        // 0.8M
  float* H0b    = ws + off; off += (size_t)BB * NCH * DI * DS;        // 0.8M

  // residual base: out = x
  {
    int n = BB * NPART * CDIM * NTOK;
    k_copy<<<(n + 255) / 256, 256, 0, stream>>>(x, out, n);
  }
  // layernorm (24576 tokens, 8 waves/block)
  k_layernorm<<<3072, 256, 0, stream>>>(x, norm_w, norm_b, normed);

  for (int m = 0; m < 2; ++m) {
    // 768 M-supertiles * 16 N-supertiles = 12288 waves / 8 per block
    k_inproj<<<1536, 256, 0, stream>>>(normed, in_proj_w, xz, m);
    for (int br = 0; br < 3; ++br) {
      k_conv<<<24576, 256, 0, stream>>>(xz, cw[br], cb[br], ubuf, br);
      k_xproj<<<576, 256, 0, stream>>>(ubuf, xw[br], dbl);              // 4608 waves
      k_dtproj<<<1536, 256, 0, stream>>>(dbl, dtw[br], dtb[br], dlt);   // 12288 waves
      k_scan1<<<dim3(NCH, DI / 16, BB), 256, 0, stream>>>(dlt, ubuf, dbl,
                                                          alog[br], Pb, Sb);
      k_scan2<<<(BB * DI * DS + 255) / 256, 256, 0, stream>>>(Pb, Sb, H0b);
      k_scan3<<<dim3(NCH, DI / 16, BB), 256, 0, stream>>>(dlt, ubuf, dbl, xz,
                                                          alog[br], dskip[br], H0b,
                                                          ysum, br, br > 0 ? 1 : 0);
    }
    // 768 * 4 = 3072 waves
    k_outproj<<<384, 256, 0, stream>>>(ysum, out_proj_w, out);
  }
}